// GNN_51049981280318
// MI455X (gfx1250) — compile-verified
//
#include <hip/hip_runtime.h>
#include <hip/hip_bf16.h>

typedef float v2f __attribute__((ext_vector_type(2)));
typedef float v8f __attribute__((ext_vector_type(8)));

#define D 64
#define LDSP 68            // LDS row pitch in floats: 68%4==0 (float4 stores), (4L+k)%64 conflict-free
#define NUM_GRAPHS 128
#define NUM_CLASSES 10
#define BN_EPS 1e-5f

// ---------------------------------------------------------------- zero fill
__global__ void gnn_zero_kernel(float* __restrict__ p, int n) {
    int i = blockIdx.x * blockDim.x + threadIdx.x;
    if (i < n) p[i] = 0.0f;
}

// ------------------------------------------------- edge scatter: segment_sum(x[src], dst)
// 16 threads per edge, each handles one float4 chunk of the 64-float row.
__global__ void gnn_edge_scatter_kernel(const float* __restrict__ x,
                                        const int* __restrict__ src,
                                        const int* __restrict__ dst,
                                        float* __restrict__ msg, int E) {
    int t = blockIdx.x * blockDim.x + threadIdx.x;
    if (t >= E * 16) return;
    int e  = t >> 4;
    int c4 = (t & 15) * 4;
    int s = src[e], d = dst[e];
    float4 v = *(const float4*)(x + (size_t)s * D + c4);
    float* o = msg + (size_t)d * D + c4;
    unsafeAtomicAdd(o + 0, v.x);
    unsafeAtomicAdd(o + 1, v.y);
    unsafeAtomicAdd(o + 2, v.z);
    unsafeAtomicAdd(o + 3, v.w);
}

// ------------------------------------------------- WMMA GEMM: h = msg@Wrel + b + x@Wroot
// 4 waves/block, each wave computes a 16x64 output tile via v_wmma_f32_16x16x4_f32.
// hout may alias msg (rows are staged to LDS before any store; dependence is row-local).
__global__ void __launch_bounds__(128)
gnn_gemm_kernel(const float* __restrict__ msg, float* __restrict__ hout,
                const float* __restrict__ x,
                const float* __restrict__ Wrel, const float* __restrict__ brel,
                const float* __restrict__ Wroot, int N) {
    __shared__ float lm[4][16 * LDSP];
    __shared__ float lx[4][16 * LDSP];
    const int wave = threadIdx.x >> 5;
    const int lane = threadIdx.x & 31;
    const int rowBase = blockIdx.x * 64 + wave * 16;

    // Stage 16 rows (64 floats each) of msg and x into LDS, coalesced float4.
    #pragma unroll
    for (int i = 0; i < 8; ++i) {
        int slot = lane + i * 32;          // 256 float4 slots
        int r  = slot >> 4;                // local row 0..15
        int c4 = (slot & 15) * 4;
        int gr = rowBase + r; if (gr > N - 1) gr = N - 1;   // clamp (stores are guarded)
        float4 vm = *(const float4*)(msg + (size_t)gr * D + c4);
        float4 vx = *(const float4*)(x   + (size_t)gr * D + c4);
        *(float4*)(&lm[wave][r * LDSP + c4]) = vm;
        *(float4*)(&lx[wave][r * LDSP + c4]) = vx;
    }
    __syncthreads();

    const int arow = lane & 15;            // A: M row
    const int kb   = (lane >> 4) * 2;      // A/B: K pair select (0 or 2)
    const int bcol = lane & 15;            // B/C/D: N col
    const float* lmw = lm[wave];
    const float* lxw = lx[wave];

    for (int ct = 0; ct < 4; ++ct) {       // 4 output col tiles of 16
        v8f acc = {};
        #pragma unroll
        for (int kk = 0; kk < D; kk += 4) {
            v2f a, b;
            // relation term: msg @ Wrel
            a[0] = lmw[arow * LDSP + kk + kb];
            a[1] = lmw[arow * LDSP + kk + kb + 1];
            b[0] = Wrel[(size_t)(kk + kb)     * D + ct * 16 + bcol];
            b[1] = Wrel[(size_t)(kk + kb + 1) * D + ct * 16 + bcol];
            acc = __builtin_amdgcn_wmma_f32_16x16x4_f32(false, a, false, b,
                                                        (short)0, acc, false, false);
            // root term: x @ Wroot
            a[0] = lxw[arow * LDSP + kk + kb];
            a[1] = lxw[arow * LDSP + kk + kb + 1];
            b[0] = Wroot[(size_t)(kk + kb)     * D + ct * 16 + bcol];
            b[1] = Wroot[(size_t)(kk + kb + 1) * D + ct * 16 + bcol];
            acc = __builtin_amdgcn_wmma_f32_16x16x4_f32(false, a, false, b,
                                                        (short)0, acc, false, false);
        }
        float bias = brel[ct * 16 + bcol];
        #pragma unroll
        for (int i = 0; i < 8; ++i) {      // C/D layout: VGPR i -> M=i (lanes<16) / i+8
            int m  = (lane < 16) ? i : i + 8;
            int gr = rowBase + m;
            if (gr < N) hout[(size_t)gr * D + ct * 16 + bcol] = acc[i] + bias;
        }
    }
}

// ------------------------------------------------- BN column stats: sum & sumsq per feature
__global__ void gnn_stats_kernel(const float* __restrict__ h, float* __restrict__ stats, int N) {
    __shared__ float ss[256], ss2[256];
    int col = threadIdx.x & 63;
    int grp = threadIdx.x >> 6;            // 0..3 row groups
    float s = 0.0f, s2 = 0.0f;
    for (int r = blockIdx.x * 4 + grp; r < N; r += gridDim.x * 4) {
        float v = h[(size_t)r * D + col];
        s += v; s2 += v * v;
    }
    ss[threadIdx.x] = s; ss2[threadIdx.x] = s2;
    __syncthreads();
    if (grp == 0) {
        s  = ss[col]  + ss[col + 64]  + ss[col + 128]  + ss[col + 192];
        s2 = ss2[col] + ss2[col + 64] + ss2[col + 128] + ss2[col + 192];
        unsafeAtomicAdd(stats + col, s);
        unsafeAtomicAdd(stats + 64 + col, s2);
    }
}

// ------------------------------------------------- BN normalize + ReLU (in place)
__global__ void gnn_bnrelu_kernel(float* __restrict__ h, const float* __restrict__ stats,
                                  const float* __restrict__ gamma, const float* __restrict__ beta,
                                  int N, float invN) {
    int t = blockIdx.x * blockDim.x + threadIdx.x;
    if (t >= N * 16) return;
    int c4 = (t & 15) * 4;
    size_t base = (size_t)(t >> 4) * D + c4;
    float4 v = *(float4*)(h + base);
    float r[4] = {v.x, v.y, v.z, v.w};
    #pragma unroll
    for (int k = 0; k < 4; ++k) {
        int c = c4 + k;
        float mean = stats[c] * invN;
        float var  = stats[64 + c] * invN - mean * mean;
        float inv  = rsqrtf(var + BN_EPS);
        r[k] = fmaxf(0.0f, (r[k] - mean) * inv * gamma[c] + beta[c]);
    }
    *(float4*)(h + base) = make_float4(r[0], r[1], r[2], r[3]);
}

// ------------------------------------------------- global mean pool (sum + counts)
__global__ void gnn_pool_kernel(const float* __restrict__ x, const int* __restrict__ batch,
                                float* __restrict__ pooled, float* __restrict__ counts, int N) {
    int t = blockIdx.x * blockDim.x + threadIdx.x;
    if (t >= N * 16) return;
    int n  = t >> 4;
    int c4 = (t & 15) * 4;
    int g = batch[n];
    float4 v = *(const float4*)(x + (size_t)n * D + c4);
    float* p = pooled + (size_t)g * D + c4;
    unsafeAtomicAdd(p + 0, v.x);
    unsafeAtomicAdd(p + 1, v.y);
    unsafeAtomicAdd(p + 2, v.z);
    unsafeAtomicAdd(p + 3, v.w);
    if (c4 == 0) unsafeAtomicAdd(counts + g, 1.0f);
}

// ------------------------------------------------- classifier: out = (pooled/cnt) @ Wcls + b
__global__ void gnn_classify_kernel(const float* __restrict__ pooled,
                                    const float* __restrict__ counts,
                                    const float* __restrict__ Wcls,
                                    const float* __restrict__ bcls,
                                    float* __restrict__ out) {
    int g = blockIdx.x;
    int c = threadIdx.x;
    if (c >= NUM_CLASSES) return;
    float inv = 1.0f / fmaxf(counts[g], 1.0f);
    float acc = 0.0f;
    #pragma unroll
    for (int d = 0; d < D; ++d) acc += pooled[(size_t)g * D + d] * Wcls[d * NUM_CLASSES + c];
    out[g * NUM_CLASSES + c] = acc * inv + bcls[c];
}

extern "C" void kernel_launch(void* const* d_in, const int* in_sizes, int n_in,
                              void* d_out, int out_size, void* d_ws, size_t ws_size,
                              hipStream_t stream) {
    const float* x_in   = (const float*)d_in[0];
    const int*   eidx   = (const int*)d_in[1];
    const int*   batch  = (const int*)d_in[2];
    const float* W_rel  = (const float*)d_in[3];   // [3,64,64]
    const float* b_rel  = (const float*)d_in[4];   // [3,64]
    const float* W_root = (const float*)d_in[5];   // [3,64,64]
    const float* gamma  = (const float*)d_in[6];   // [3,64]
    const float* beta   = (const float*)d_in[7];   // [3,64]
    const float* W_cls  = (const float*)d_in[8];   // [64,10]
    const float* b_cls  = (const float*)d_in[9];   // [10]
    float* out = (float*)d_out;

    const int N = in_sizes[0] / D;
    const int E = in_sizes[1] / 2;
    const int* src = eidx;
    const int* dst = eidx + E;

    float* buf0   = (float*)d_ws;
    float* buf1   = buf0 + (size_t)N * D;
    float* stats  = buf1 + (size_t)N * D;          // 128 floats
    float* pooled = stats + 128;                   // 128*64
    float* counts = pooled + NUM_GRAPHS * D;       // 128

    float* bufs[2] = {buf0, buf1};
    const float invN = 1.0f / (float)N;
    const int ND = N * D;

    const float* xcur = x_in;
    for (int l = 0; l < 3; ++l) {
        float* work = bufs[l & 1];                 // msg -> h -> x_next, all in place
        gnn_zero_kernel<<<(ND + 255) / 256, 256, 0, stream>>>(work, ND);
        gnn_zero_kernel<<<1, 128, 0, stream>>>(stats, 128);
        gnn_edge_scatter_kernel<<<(E * 16 + 255) / 256, 256, 0, stream>>>(xcur, src, dst, work, E);
        gnn_gemm_kernel<<<(N + 63) / 64, 128, 0, stream>>>(
            work, work, xcur,
            W_rel + (size_t)l * D * D, b_rel + (size_t)l * D,
            W_root + (size_t)l * D * D, N);
        gnn_stats_kernel<<<256, 256, 0, stream>>>(work, stats, N);
        gnn_bnrelu_kernel<<<(N * 16 + 255) / 256, 256, 0, stream>>>(
            work, stats, gamma + (size_t)l * D, beta + (size_t)l * D, N, invN);
        xcur = work;
    }

    gnn_zero_kernel<<<(NUM_GRAPHS * D + NUM_GRAPHS + 255) / 256, 256, 0, stream>>>(
        pooled, NUM_GRAPHS * D + NUM_GRAPHS);      // pooled + counts contiguous
    gnn_pool_kernel<<<(N * 16 + 255) / 256, 256, 0, stream>>>(xcur, batch, pooled, counts, N);
    gnn_classify_kernel<<<NUM_GRAPHS, 32, 0, stream>>>(pooled, counts, W_cls, b_cls, out);
}